// GaussianEncoderLayer_7533372637347
// MI455X (gfx1250) — compile-verified
//
#include <hip/hip_runtime.h>
#include <hip/hip_bf16.h>
#include <math.h>

typedef __attribute__((ext_vector_type(2))) float v2f;
typedef __attribute__((ext_vector_type(8))) float v8f;

#define C_DIM 128
#define HID   256
#define ROWS  4096   // B*G
// workspace layout (float offsets)
#define WS_T  0ull          // transposed features, 5,744,640 floats
#define WS_H  5744640ull    // hidden 4096x256
#define WS_X  6793216ull    // xfeat  4096x128

__device__ __forceinline__ float sigmoidf_(float x) { return 1.f / (1.f + expf(-x)); }

// ---------------- feature transpose: (BN,C,HW) -> (BN,HW,C) ----------------
__global__ void k_transpose(const float* __restrict__ in, float* __restrict__ out,
                            int HW, int total) {
  int tid = blockIdx.x * blockDim.x + threadIdx.x;
  if (tid >= total) return;
  int c = tid & (C_DIM - 1);
  int rest = tid >> 7;
  int s = rest % HW;
  int bn = rest / HW;
  out[tid] = in[((size_t)bn * C_DIM + c) * HW + s];
}

// ---------------- GEMM1: (4096x2)@(2x256)+b1, relu -> H (f32 WMMA) ----------------
__global__ __launch_bounds__(256) void k_mlp1(const float* __restrict__ gaus,
                                              const float* __restrict__ w1,
                                              const float* __restrict__ b1,
                                              float* __restrict__ H) {
  int wave = (blockIdx.x * 256 + threadIdx.x) >> 5;   // 0..4095
  int lane = threadIdx.x & 31;
  int l = lane & 15, hi = lane >> 4;
  int row0 = (wave >> 4) * 16;   // 256 row tiles
  int col0 = (wave & 15) * 16;   // 16 col tiles
  // A 16x4 (K=2 real, K=2..3 zero): lanes0-15 hold K0,K1; lanes16-31 hold K2,K3
  v2f a; a.x = 0.f; a.y = 0.f;
  v2f bv; bv.x = 0.f; bv.y = 0.f;
  if (!hi) {
    const float* gp = gaus + (size_t)(row0 + l) * 10;
    a.x = gp[0]; a.y = gp[1];
    bv.x = w1[col0 + l];          // K=0 row of w1
    bv.y = w1[HID + col0 + l];    // K=1 row of w1
  }
  v8f c = {};
  c = __builtin_amdgcn_wmma_f32_16x16x4_f32(false, a, false, bv, (short)0, c, false, false);
  #pragma unroll
  for (int r = 0; r < 8; ++r) {
    int row = row0 + r + hi * 8;
    int col = col0 + l;
    float h = c[r] + b1[col];
    H[(size_t)row * HID + col] = fmaxf(h, 0.f);
  }
}

// ---------------- GEMM2: (4096x256)@(256x128)+b2, + gaussians_feat -> xfeat ----------------
__global__ __launch_bounds__(256) void k_mlp2(const float* __restrict__ H,
                                              const float* __restrict__ w2,
                                              const float* __restrict__ b2,
                                              const float* __restrict__ gfeat,
                                              float* __restrict__ X) {
  int wave = (blockIdx.x * 256 + threadIdx.x) >> 5;   // 0..2047
  int lane = threadIdx.x & 31;
  int l = lane & 15, hi = lane >> 4;
  int row0 = (wave >> 3) * 16;   // 256 row tiles
  int col0 = (wave & 7) * 16;    // 8 col tiles
  v8f c = {};
  const float* hrow = H + (size_t)(row0 + l) * HID;
  #pragma unroll 4
  for (int k0 = 0; k0 < HID; k0 += 4) {
    v2f a, bv;
    a.x = hrow[k0 + 2 * hi];
    a.y = hrow[k0 + 2 * hi + 1];
    bv.x = w2[(size_t)(k0 + 2 * hi) * C_DIM + col0 + l];
    bv.y = w2[(size_t)(k0 + 2 * hi + 1) * C_DIM + col0 + l];
    c = __builtin_amdgcn_wmma_f32_16x16x4_f32(false, a, false, bv, (short)0, c, false, false);
  }
  #pragma unroll
  for (int r = 0; r < 8; ++r) {
    int row = row0 + r + hi * 8;
    int col = col0 + l;
    float pe = c[r] + b2[col];
    X[(size_t)row * C_DIM + col] = gfeat[(size_t)row * C_DIM + col] + pe;
  }
}

// ---------------- sampler + camera reduce + LayerNorm (one gaussian per block) ----------------
__global__ __launch_bounds__(128) void k_sample_ln(
    const float* __restrict__ T,      // transposed features (all levels)
    const float* __restrict__ l2i,    // (12,4,4)
    const float* __restrict__ gaus,   // (4096,10)
    const float* __restrict__ eps,    // (4096,16,3)
    const float* __restrict__ xfeat,  // (4096,128)
    const float* __restrict__ ln_g, const float* __restrict__ ln_b,
    float* __restrict__ out,          // (4096,128)
    float* __restrict__ out_pts)      // (4096,16,3)
{
  __shared__ int4 s_idx[384];
  __shared__ float4 s_w[384];
  __shared__ float s_red[128];
  __shared__ float s_mu, s_rstd;

  const int row = blockIdx.x;      // b*2048+g
  const int b = row >> 11;
  const int tid = threadIdx.x;

  if (tid < 96) {                  // geometry: (n,p) = (tid/16, tid%16)
    const int n = tid >> 4, p = tid & 15;
    const float* gp = gaus + (size_t)row * 10;
    float mx = fmaf(gp[0], 100.f, -50.f);
    float my = fmaf(gp[1], 100.f, -50.f);
    float mz = fmaf(gp[2], 8.f, -5.f);
    float s0 = fmaf(sigmoidf_(gp[3]), 3.9f, 0.1f);
    float s1 = fmaf(sigmoidf_(gp[4]), 3.9f, 0.1f);
    float s2 = fmaf(sigmoidf_(gp[5]), 3.9f, 0.1f);
    float qw = gp[6], qx = gp[7], qy = gp[8], qz = gp[9];
    float qn = fmaxf(sqrtf(qw*qw + qx*qx + qy*qy + qz*qz), 1e-12f);
    float inv = 1.f / qn; qw *= inv; qx *= inv; qy *= inv; qz *= inv;
    float R00 = 1.f - 2.f*(qy*qy + qz*qz), R01 = 2.f*(qx*qy - qz*qw), R02 = 2.f*(qx*qz + qy*qw);
    float R10 = 2.f*(qx*qy + qz*qw), R11 = 1.f - 2.f*(qx*qx + qz*qz), R12 = 2.f*(qy*qz - qx*qw);
    float R20 = 2.f*(qx*qz - qy*qw), R21 = 2.f*(qy*qz + qx*qw), R22 = 1.f - 2.f*(qx*qx + qy*qy);
    const float* ep = eps + ((size_t)row * 16 + p) * 3;
    float e0 = ep[0], e1 = ep[1], e2 = ep[2];
    float px = mx + R00*s0*e0 + R01*s1*e1 + R02*s2*e2;
    float py = my + R10*s0*e0 + R11*s1*e1 + R12*s2*e2;
    float pz = mz + R20*s0*e0 + R21*s1*e1 + R22*s2*e2;
    if (n == 0) {
      float* op = out_pts + ((size_t)row * 16 + p) * 3;
      op[0] = px; op[1] = py; op[2] = pz;
    }
    const float* M = l2i + (size_t)(b * 6 + n) * 16;
    float c0 = M[0]*px + M[1]*py + M[2]*pz  + M[3];
    float c1 = M[4]*px + M[5]*py + M[6]*pz  + M[7];
    float c2 = M[8]*px + M[9]*py + M[10]*pz + M[11];
    float valid = (c2 > 1e-5f) ? 0.25f : 0.f;   // fold 1/num_levels in
    float zc = fmaxf(c2, 1e-5f);
    float u = c0 / (zc * 704.f);
    float v = c1 / (zc * 256.f);
    const int LH[4]   = {32, 16, 8, 4};
    const int LW[4]   = {88, 44, 22, 11};
    const int LOFF[4] = {0, 4325376, 5406720, 5677056};
    #pragma unroll
    for (int lv = 0; lv < 4; ++lv) {
      int Hl = LH[lv], Wl = LW[lv];
      float x = u * (float)Wl - 0.5f;
      float y = v * (float)Hl - 0.5f;
      float x0f = floorf(x), y0f = floorf(y);
      int x0 = (int)x0f, y0 = (int)y0f;
      float wx1 = x - x0f, wx0 = 1.f - wx1;
      float wy1 = y - y0f, wy0 = 1.f - wy1;
      int base = LOFF[lv] + (b * 6 + n) * Hl * Wl * C_DIM;
      int4 idx; float4 w;
      {
        int xi = x0, yi = y0;
        bool inb = (xi >= 0) & (xi < Wl) & (yi >= 0) & (yi < Hl);
        int xc = min(max(xi, 0), Wl - 1), yc = min(max(yi, 0), Hl - 1);
        idx.x = base + (yc * Wl + xc) * C_DIM; w.x = inb ? wx0 * wy0 * valid : 0.f;
      }
      {
        int xi = x0 + 1, yi = y0;
        bool inb = (xi >= 0) & (xi < Wl) & (yi >= 0) & (yi < Hl);
        int xc = min(max(xi, 0), Wl - 1), yc = min(max(yi, 0), Hl - 1);
        idx.y = base + (yc * Wl + xc) * C_DIM; w.y = inb ? wx1 * wy0 * valid : 0.f;
      }
      {
        int xi = x0, yi = y0 + 1;
        bool inb = (xi >= 0) & (xi < Wl) & (yi >= 0) & (yi < Hl);
        int xc = min(max(xi, 0), Wl - 1), yc = min(max(yi, 0), Hl - 1);
        idx.z = base + (yc * Wl + xc) * C_DIM; w.z = inb ? wx0 * wy1 * valid : 0.f;
      }
      {
        int xi = x0 + 1, yi = y0 + 1;
        bool inb = (xi >= 0) & (xi < Wl) & (yi >= 0) & (yi < Hl);
        int xc = min(max(xi, 0), Wl - 1), yc = min(max(yi, 0), Hl - 1);
        idx.w = base + (yc * Wl + xc) * C_DIM; w.w = inb ? wx1 * wy1 * valid : 0.f;
      }
      s_idx[tid * 4 + lv] = idx;
      s_w[tid * 4 + lv] = w;
    }
  }
  __syncthreads();

  // each thread owns one channel; stream all (sample, level) records
  float acc = 0.f;
  const int c = tid;
  for (int e = 0; e < 384; ++e) {
    float4 w = s_w[e];
    if ((w.x == 0.f) & (w.y == 0.f) & (w.z == 0.f) & (w.w == 0.f)) continue; // block-uniform skip
    int4 ix = s_idx[e];
    acc = fmaf(w.x, T[(size_t)ix.x + c], acc);
    acc = fmaf(w.y, T[(size_t)ix.y + c], acc);
    acc = fmaf(w.z, T[(size_t)ix.z + c], acc);
    acc = fmaf(w.w, T[(size_t)ix.w + c], acc);
  }

  float x = xfeat[(size_t)row * C_DIM + c] + acc;
  // LayerNorm over 128 channels
  s_red[tid] = x; __syncthreads();
  for (int s = 64; s > 0; s >>= 1) { if (tid < s) s_red[tid] += s_red[tid + s]; __syncthreads(); }
  if (tid == 0) s_mu = s_red[0] * (1.f / 128.f);
  __syncthreads();
  float mu = s_mu;
  float d = x - mu;
  s_red[tid] = d * d; __syncthreads();
  for (int s = 64; s > 0; s >>= 1) { if (tid < s) s_red[tid] += s_red[tid + s]; __syncthreads(); }
  if (tid == 0) s_rstd = rsqrtf(s_red[0] * (1.f / 128.f) + 1e-5f);
  __syncthreads();
  out[(size_t)row * C_DIM + c] = d * s_rstd * ln_g[c] + ln_b[c];
}

extern "C" void kernel_launch(void* const* d_in, const int* in_sizes, int n_in,
                              void* d_out, int out_size, void* d_ws, size_t ws_size,
                              hipStream_t stream) {
  const float* feat[4] = {(const float*)d_in[0], (const float*)d_in[1],
                          (const float*)d_in[2], (const float*)d_in[3]};
  const float* l2i   = (const float*)d_in[4];
  const float* gaus  = (const float*)d_in[5];
  const float* gfeat = (const float*)d_in[6];
  const float* eps   = (const float*)d_in[7];
  const float* w1    = (const float*)d_in[8];
  const float* b1    = (const float*)d_in[9];
  const float* w2    = (const float*)d_in[10];
  const float* b2    = (const float*)d_in[11];
  const float* ln_g  = (const float*)d_in[12];
  const float* ln_b  = (const float*)d_in[13];

  float* ws = (float*)d_ws;
  float* T = ws + WS_T;
  float* H = ws + WS_H;
  float* X = ws + WS_X;
  float* out = (float*)d_out;
  float* out_pts = out + (size_t)ROWS * C_DIM;

  // 1) transpose features to channel-last (coalesced gathers later)
  const int    HWs[4]  = {2816, 704, 176, 44};
  const size_t Toff[4] = {0, 4325376, 5406720, 5677056};
  for (int lv = 0; lv < 4; ++lv) {
    int total = 12 * HWs[lv] * C_DIM;
    k_transpose<<<(total + 255) / 256, 256, 0, stream>>>(feat[lv], T + Toff[lv], HWs[lv], total);
  }
  // 2) MLP via f32 WMMA
  k_mlp1<<<512, 256, 0, stream>>>(gaus, w1, b1, H);       // 4096 waves, 1 WMMA each
  k_mlp2<<<256, 256, 0, stream>>>(H, w2, b2, gfeat, X);   // 2048 waves, 64 WMMA each
  // 3) gather + reduce + LayerNorm
  k_sample_ln<<<ROWS, 128, 0, stream>>>(T, l2i, gaus, eps, X, ln_g, ln_b, out, out_pts);
}